// GC_41455024340992
// MI455X (gfx1250) — compile-verified
//
#include <hip/hip_runtime.h>

// GCN layer, restructured:
//   d[i]    = 1/sqrt(1 + rowsum(adj, i))
//   w2      = pair-interleaved copy of kernel            (kernel 1b)
//   y2[j,:] = d[j] * (features @ kernel)[j,:]            (kernel 2, WMMA f32, pair-interleaved)
//   out[g,:]= relu(d[i]*(adj[i,:] @ y + y[i,:]) + bias), i = gather[g]  (kernel 3, WMMA f32)
// Pair-interleaved layout for B operands: elem (j,n) at ((j>>1)*256 + n)*2 + (j&1).
// -> async b128 global->LDS copies stage it unchanged, and each WMMA B fragment
//    is a single bank-conflict-free ds_load_b64 with an immediate offset.

typedef __attribute__((ext_vector_type(2))) float v2f;
typedef __attribute__((ext_vector_type(8))) float v8f;

#define N_NODES 8192
#define F_DIM   512
#define UNITS   256
#define N_GATHER 4096

#define A_ST 36    // 32 K + pad; rows 144B (16B aligned); 36r mod 64 -> 16 distinct banks
#define PW2  544   // pair-row stride, N=256: 2*256+32; 544 % 64 == 32 -> hi half-wave on other banks
#define PW3  288   // pair-row stride, N=128: 2*128+32; 288 % 64 == 32

#define USE_ASYNC 1

__device__ __forceinline__ unsigned to_lds(const void* p) {
    return (unsigned)(size_t)(__attribute__((address_space(3))) const void*)p;
}

// 16B global -> LDS copy. Async (ASYNCcnt-tracked) on the CDNA5 path.
__device__ __forceinline__ void copy16(const float* __restrict__ g, float* l) {
#if USE_ASYNC
    asm volatile("global_load_async_to_lds_b128 %0, %1, off"
                 :: "v"(to_lds(l)), "v"((unsigned long long)(size_t)g)
                 : "memory");
#else
    *(float4*)l = *(const float4*)g;
#endif
}

__device__ __forceinline__ void wait_async() {
#if USE_ASYNC
    asm volatile("s_wait_asynccnt 0x0" ::: "memory");
#endif
}

// ---------------- Kernel 1: d[i] = 1/sqrt(1 + sum_j adj[i,j]) ----------------
__global__ __launch_bounds__(256) void gcn_rowsum(const float* __restrict__ adj,
                                                  float* __restrict__ d) {
    __shared__ float red[256];
    const int row = blockIdx.x;
    const int t = threadIdx.x;
    const float4* rowp = (const float4*)(adj + (size_t)row * N_NODES);
    float s = 0.f;
#pragma unroll
    for (int k = 0; k < 8; ++k) {
        float4 v = rowp[t + k * 256];
        s += (v.x + v.y) + (v.z + v.w);
    }
    red[t] = s;
    __syncthreads();
    for (int off = 128; off > 0; off >>= 1) {
        if (t < off) red[t] += red[t + off];
        __syncthreads();
    }
    if (t == 0) d[row] = 1.0f / sqrtf(red[0] + 1.0f);
}

// ---------------- Kernel 1b: w2 = pair-interleave(kernel) ----------------
// w2[(jp*256 + n)*2 + par] = wmat[2*jp + par][n]
__global__ __launch_bounds__(256) void gcn_pairw(const float* __restrict__ wmat,
                                                 float* __restrict__ w2) {
    const int i = blockIdx.x * 256 + threadIdx.x;   // 32768 chunks of 16B
    const int jp = i >> 7;
    const int n2 = (i & 127) * 2;
    float2 lo = *(const float2*)(wmat + (size_t)(2 * jp) * UNITS + n2);
    float2 hi = *(const float2*)(wmat + (size_t)(2 * jp + 1) * UNITS + n2);
    float4 o;
    o.x = lo.x; o.y = hi.x; o.z = lo.y; o.w = hi.y;
    *(float4*)(w2 + ((size_t)jp * UNITS + n2) * 2) = o;
}

// ---------------- Kernel 2: y2 = d * (features @ kernel), pair-interleaved ----------------
// 256 blocks x 512 threads (16 waves). 32 M-rows, 256 N per block.
__global__ __launch_bounds__(512) void gcn_xw(const float* __restrict__ features,
                                              const float* __restrict__ w2,
                                              const float* __restrict__ d,
                                              float* __restrict__ y2) {
    __shared__ alignas(16) float As[2][32 * A_ST];
    __shared__ alignas(16) float Bs[2][16 * PW2];
    const int t = threadIdx.x;
    const int lane = t & 31;
    const int wave = t >> 5;
    const int mBase = blockIdx.x * 32;
    const int mt  = wave >> 3;
    const int nt0 = (wave & 7) * 2;
    const int hi = (lane >= 16) ? 1 : 0;
    const int mrow = lane & 15;

    v8f acc0 = {};
    v8f acc1 = {};

    auto issue = [&](int buf, int kBase) {
        if (t < 256) {                       // A tile: 32 x 32 = 256 x 16B
            const int r  = t >> 3;
            const int c4 = (t & 7) * 4;
            copy16(features + (size_t)(mBase + r) * F_DIM + kBase + c4,
                   &As[buf][r * A_ST + c4]);
        }
        const int kp0 = kBase >> 1;
#pragma unroll
        for (int it = 0; it < 4; ++it) {     // B tile: 16 jp-rows x 512 = 2048 x 16B
            const int i = t + it * 512;
            const int jp = i >> 7;
            const int c4 = (i & 127) * 4;
            copy16(w2 + (size_t)(kp0 + jp) * (UNITS * 2) + c4,
                   &Bs[buf][jp * PW2 + c4]);
        }
    };

    issue(0, 0);
    const int NCHUNK = F_DIM / 32;
    for (int c = 0; c < NCHUNK; ++c) {
        const int cur = c & 1;
        wait_async();
        __syncthreads();                     // chunk c resident; buf cur^1 free
        if (c + 1 < NCHUNK) issue(cur ^ 1, (c + 1) * 32);
        const float* Abase = &As[cur][(mt * 16 + mrow) * A_ST + hi * 2];
        const float* Bbase = &Bs[cur][(nt0 * 16 + mrow) * 2 + hi * PW2];
#pragma unroll
        for (int kk = 0; kk < 8; ++kk) {
            v2f a  = *(const v2f*)(Abase + kk * 4);
            v2f b0 = *(const v2f*)(Bbase + kk * 2 * PW2);
            v2f b1 = *(const v2f*)(Bbase + kk * 2 * PW2 + 32);
            acc0 = __builtin_amdgcn_wmma_f32_16x16x4_f32(false, a, false, b0, (short)0, acc0, false, false);
            acc1 = __builtin_amdgcn_wmma_f32_16x16x4_f32(false, a, false, b1, (short)0, acc1, false, false);
        }
    }

    const int col0 = nt0 * 16 + mrow;
    const int col1 = (nt0 + 1) * 16 + mrow;
#pragma unroll
    for (int r = 0; r < 8; ++r) {
        const int grow = mBase + mt * 16 + r + hi * 8;
        const float dv = d[grow];
        const size_t base = ((size_t)(grow >> 1) * UNITS) * 2 + (grow & 1);
        y2[base + col0 * 2] = dv * acc0[r];
        y2[base + col1 * 2] = dv * acc1[r];
    }
}

// ---------------- Kernel 3: out[g,:] = relu(d[i]*(adj[i,:]@y + y[i,:]) + bias) ----------------
// Grid (128,2): 32 gather-rows x 128 N per block; 512 threads = 16 waves, one tile each.
__global__ __launch_bounds__(512) void gcn_aggr(const float* __restrict__ adj,
                                                const int* __restrict__ gather,
                                                const float* __restrict__ y2,
                                                const float* __restrict__ d,
                                                const float* __restrict__ bias,
                                                float* __restrict__ out) {
    __shared__ alignas(16) float As[2][32 * A_ST];
    __shared__ alignas(16) float Bs[2][16 * PW3];
    __shared__ int gidx[32];
    const int t = threadIdx.x;
    const int lane = t & 31;
    const int wave = t >> 5;
    const int gBase = blockIdx.x * 32;
    const int nBase = blockIdx.y * 128;
    const int mt = wave >> 3;   // 0..1
    const int nt = wave & 7;    // 0..7
    const int hi = (lane >= 16) ? 1 : 0;
    const int mrow = lane & 15;

    if (t < 32) gidx[t] = gather[gBase + t];
    __syncthreads();            // gidx visible before async A issues read it

    v8f acc = {};

    auto issue = [&](int buf, int kBase) {
        if (t < 256) {                       // A tile: 32 gathered rows x 32 K
            const int r  = t >> 3;
            const int c4 = (t & 7) * 4;
            copy16(adj + (size_t)gidx[r] * N_NODES + kBase + c4,
                   &As[buf][r * A_ST + c4]);
        }
        const int kp0 = kBase >> 1;
#pragma unroll
        for (int it = 0; it < 2; ++it) {     // B tile: 16 jp-rows x 256 = 1024 x 16B (y2 in L2)
            const int i = t + it * 512;
            const int jp = i >> 6;
            const int c4 = (i & 63) * 4;
            copy16(y2 + ((size_t)(kp0 + jp) * UNITS + nBase) * 2 + c4,
                   &Bs[buf][jp * PW3 + c4]);
        }
    };

    issue(0, 0);
    const int NCHUNK = N_NODES / 32;
    for (int c = 0; c < NCHUNK; ++c) {
        const int cur = c & 1;
        wait_async();
        __syncthreads();
        if (c + 1 < NCHUNK) issue(cur ^ 1, (c + 1) * 32);
        const float* Abase = &As[cur][(mt * 16 + mrow) * A_ST + hi * 2];
        const float* Bbase = &Bs[cur][(nt * 16 + mrow) * 2 + hi * PW3];
#pragma unroll
        for (int kk = 0; kk < 8; ++kk) {
            v2f a = *(const v2f*)(Abase + kk * 4);
            v2f b = *(const v2f*)(Bbase + kk * 2 * PW3);
            acc = __builtin_amdgcn_wmma_f32_16x16x4_f32(false, a, false, b, (short)0, acc, false, false);
        }
    }

    const int col = nBase + nt * 16 + mrow;
    const float bv = bias[col];
#pragma unroll
    for (int r = 0; r < 8; ++r) {
        const int row = mt * 16 + r + hi * 8;
        const int gi  = gidx[row];
        const float dv = d[gi];
        const float self = y2[((size_t)(gi >> 1) * UNITS + col) * 2 + (gi & 1)];
        float v = dv * (acc[r] + self) + bv;
        out[(size_t)(gBase + row) * UNITS + col] = v > 0.f ? v : 0.f;
    }
}

// ---------------- Launch ----------------
extern "C" void kernel_launch(void* const* d_in, const int* in_sizes, int n_in,
                              void* d_out, int out_size, void* d_ws, size_t ws_size,
                              hipStream_t stream) {
    const float* adj      = (const float*)d_in[0];
    const int*   gather   = (const int*)d_in[1];
    const float* features = (const float*)d_in[2];
    const float* wmat     = (const float*)d_in[3];
    const float* bias     = (const float*)d_in[4];
    float* out = (float*)d_out;

    // Workspace: d (32KB) | w2 (512KB, pair-interleaved kernel) | y2 (8MB, pair-interleaved)
    float* dvec = (float*)d_ws;
    float* w2   = (float*)((char*)d_ws + 32768);
    float* y2   = (float*)((char*)d_ws + 32768 + 524288);

    gcn_rowsum<<<N_NODES, 256, 0, stream>>>(adj, dvec);
    gcn_pairw<<<(F_DIM / 2) * (UNITS / 2) / 256, 256, 0, stream>>>(wmat, w2);
    gcn_xw<<<N_NODES / 32, 512, 0, stream>>>(features, w2, dvec, y2);
    dim3 g3(N_GATHER / 32, 2);
    gcn_aggr<<<g3, 512, 0, stream>>>(adj, gather, y2, dvec, bias, out);
}